// ChiSquareMatchingLoss_82600811036703
// MI455X (gfx1250) — compile-verified
//
#include <hip/hip_runtime.h>
#include <hip/hip_bf16.h>

// ChiSquareMatchingLoss: soft_sort (bitonic sort + PAV isotonic regression) +
// max-abs-diff, single workgroup resident in LDS on one WGP (CDNA5 / gfx1250).
//
// d_in[0] = x      (float32, N=4096)
// d_in[1] = i_seq  (float32, N=4096)
// d_out   = scalar float32

#define NELEM 4096
#define BLOCK 1024
#define EPS_F 0.1f

#define ASG __attribute__((address_space(1)))
#define ASL __attribute__((address_space(3)))

typedef int v4i __attribute__((ext_vector_type(4)));
typedef ASG v4i* gv4i_ptr;  // global 16-byte vector pointer
typedef ASL v4i* lv4i_ptr;  // LDS 16-byte vector pointer

#if __has_builtin(__builtin_amdgcn_global_load_async_to_lds_b128)
#define HAVE_ASYNC_LDS 1
#else
#define HAVE_ASYNC_LDS 0
#warning "gfx1250 async-to-LDS builtin unavailable; using VALU fallback copy"
#endif

__global__ __launch_bounds__(BLOCK, 1) void chisq_softsort_kernel(
    const float* __restrict__ x, const float* __restrict__ iseq,
    float* __restrict__ out) {
  __shared__ float a[NELEM];       // x -> sorted -> y -> sol (reused in place)
  __shared__ float s_mean[NELEM];  // PAV block-mean stack; reused for reduction
  __shared__ int s_start[NELEM];   // PAV block-start stack

  const int tid = threadIdx.x;

  // ---- Phase 1: stage x into LDS via CDNA5 async global->LDS DMA path ----
#if HAVE_ASYNC_LDS
  // Each of the 1024 lanes issues one 16-byte async transfer (4096 floats total).
  __builtin_amdgcn_global_load_async_to_lds_b128(
      (gv4i_ptr)(x + 4 * tid), (lv4i_ptr)(&a[4 * tid]), /*offset=*/0,
      /*cpol=*/0);
#if __has_builtin(__builtin_amdgcn_s_wait_asynccnt)
  __builtin_amdgcn_s_wait_asynccnt(0);
#else
  asm volatile("s_wait_asynccnt 0" ::: "memory");
#endif
#else
  for (int i = tid; i < NELEM; i += BLOCK) a[i] = x[i];
#endif
  __syncthreads();

  // ---- Phase 2: bitonic sort ascending, fully in LDS ----
  for (unsigned k = 2; k <= NELEM; k <<= 1) {
    for (unsigned j = k >> 1; j > 0; j >>= 1) {
      __syncthreads();
      for (unsigned t = tid; t < NELEM; t += BLOCK) {
        unsigned p = t ^ j;
        if (p > t) {
          float v0 = a[t];
          float v1 = a[p];
          bool up = ((t & k) == 0);
          if ((v0 > v1) == up) {  // swap when out of order for this direction
            a[t] = v1;
            a[p] = v0;
          }
        }
      }
    }
  }
  __syncthreads();

  // ---- Phase 3: y_i = w_i + x_asc[i],  w_i = (N - i)/eps  (in place) ----
  for (int i = tid; i < NELEM; i += BLOCK) {
    float w = (float)(NELEM - i) / EPS_F;
    a[i] = w + a[i];
  }
  __syncthreads();

  // ---- Phase 4: PAV isotonic regression (non-increasing fit of y) ----
  // Nearly no merges occur (w spacing = 10 >> sorted-normal gaps), so this
  // single-threaded pass is ~O(N) LDS ops. Equivalent to the reference's
  // minimax formula (block averaging).
  if (tid == 0) {
    int nb = 0;
    for (int i = 0; i < NELEM; ++i) {
      float m = a[i];
      float cnt = 1.0f;
      int st = i;
      // non-increasing requires mean[prev] >= mean[cur]; merge on violation
      while (nb > 0 && m > s_mean[nb - 1]) {
        float cprev = (float)(st - s_start[nb - 1]);
        m = (m * cnt + s_mean[nb - 1] * cprev) / (cnt + cprev);
        cnt += cprev;
        st = s_start[nb - 1];
        --nb;
      }
      s_mean[nb] = m;
      s_start[nb] = st;
      ++nb;
    }
    // Expand block means back into a[] (a[i] = sol_i)
    for (int b = 0; b < nb; ++b) {
      int e = (b + 1 < nb) ? s_start[b + 1] : NELEM;
      float m = s_mean[b];
      for (int i = s_start[b]; i < e; ++i) a[i] = m;
    }
  }
  __syncthreads();

  // ---- Phase 5: max_i | i_seq[i] - (sol_i - w_i) |, LDS tree reduction ----
  float lm = 0.0f;
  for (int i = tid; i < NELEM; i += BLOCK) {
    float w = (float)(NELEM - i) / EPS_F;
    float xs = a[i] - w;  // soft-sorted value (ascending)
    lm = fmaxf(lm, fabsf(iseq[i] - xs));
  }
  s_mean[tid] = lm;
  __syncthreads();
  for (int s = BLOCK >> 1; s > 0; s >>= 1) {
    if (tid < s) s_mean[tid] = fmaxf(s_mean[tid], s_mean[tid + s]);
    __syncthreads();
  }
  if (tid == 0) out[0] = s_mean[0];
}

extern "C" void kernel_launch(void* const* d_in, const int* in_sizes, int n_in,
                              void* d_out, int out_size, void* d_ws,
                              size_t ws_size, hipStream_t stream) {
  (void)in_sizes;
  (void)n_in;
  (void)out_size;
  (void)d_ws;
  (void)ws_size;
  const float* x = (const float*)d_in[0];
  const float* iseq = (const float*)d_in[1];
  float* out = (float*)d_out;
  chisq_softsort_kernel<<<dim3(1), dim3(BLOCK), 0, stream>>>(x, iseq, out);
}